// SpatialHRVQTokenizer_15908558865671
// MI455X (gfx1250) — compile-verified
//
#include <hip/hip_runtime.h>
#include <stdint.h>

#define D       384
#define KCODES  256
#define TOKPB   64    // tokens per block
#define XPAD    388   // X tile row stride (floats): 16B aligned, conflict-free (4m mod 64 distinct)
#define CHUNK   16    // k-values per staged codebook chunk
#define CPAD    20    // chunk row stride (floats): 16B aligned, conflict-free (20m mod 64 distinct)
#define NCHUNK  24    // 384 / 16
#define CBUF    (KCODES * CPAD)   // floats per chunk buffer

typedef float v2f __attribute__((ext_vector_type(2)));
typedef float v8f __attribute__((ext_vector_type(8)));

// ---- CDNA5 async global->LDS (ASYNCcnt-tracked), per CDNA5 ISA ----
__device__ __forceinline__ void async_ld_b128(uint32_t lds_off, const float* g) {
    asm volatile("global_load_async_to_lds_b128 %0, %1, off"
                 :: "v"(lds_off), "v"((uint64_t)(uintptr_t)g)
                 : "memory");
}
// async loads complete in order: waiting <=4 means everything before the last 4 has landed
__device__ __forceinline__ void wait_async_le4() {
    asm volatile("s_wait_asynccnt 4" ::: "memory");
}
__device__ __forceinline__ void wait_async_le0() {
    asm volatile("s_wait_asynccnt 0" ::: "memory");
}

__global__ void vq_norms_kernel(const float* __restrict__ cb0,
                                const float* __restrict__ cb1,
                                const float* __restrict__ cb2,
                                float* __restrict__ norms) {
    const float* cb = (blockIdx.x == 0) ? cb0 : (blockIdx.x == 1) ? cb1 : cb2;
    int k = threadIdx.x;
    float s = 0.f;
    for (int i = 0; i < D; ++i) { float v = cb[k * D + i]; s += v * v; }
    norms[blockIdx.x * KCODES + k] = s;
}

__global__ void vq_zero_loss_kernel(float* loss) { *loss = 0.f; }

__global__ __launch_bounds__(256)
void vq_level_kernel(const float* __restrict__ X, const float* __restrict__ CB,
                     const float* __restrict__ cbn,
                     float* __restrict__ Q, float* __restrict__ IDXF,
                     float* __restrict__ loss_out, float beta_scale) {
    __shared__ float xs[TOKPB * XPAD];   // ~97 KB
    __shared__ float cbs[2 * CBUF];      // ~40 KB (double-buffered codebook chunks)
    __shared__ float xnorm[TOKPB];
    __shared__ float redv[2 * TOKPB];
    __shared__ int   redi[2 * TOKPB];
    __shared__ int   tokidx[TOKPB];
    __shared__ float lossred[8];

    const int tid  = threadIdx.x;
    const int lane = tid & 31;
    const int wave = tid >> 5;
    const int half     = wave >> 1;  // token group: tokens [half*16, half*16+16)
    const int codehalf = wave & 1;   // code group: codes [codehalf*128, +128) as 8 N-tiles
    const long tok0 = (long)blockIdx.x * TOKPB;

    // ---- prologue: async-stage X tile (64 rows x 96 float4 = 24/thread) + chunk 0 ----
    {
        const float* xbase = X + tok0 * D;
        #pragma unroll
        for (int it = 0; it < 24; ++it) {
            int idx = it * 256 + tid;
            int row = idx / 96;
            int c4  = idx % 96;
            async_ld_b128((uint32_t)(uintptr_t)&xs[row * XPAD + c4 * 4],
                          xbase + row * D + c4 * 4);
        }
        #pragma unroll
        for (int it = 0; it < 4; ++it) {
            int idx = it * 256 + tid;
            int n   = idx >> 2;
            int c4  = idx & 3;
            async_ld_b128((uint32_t)(uintptr_t)&cbs[n * CPAD + c4 * 4],
                          CB + n * D + c4 * 4);
        }
    }

    // ---- double-buffered WMMA accumulation of X . CB^T over D ----
    v8f acc[8] = {};
    const int m  = lane & 15;        // A: row M / B: col N (lanes 0-15 vs 16-31)
    const int ks = (lane >> 4) * 2;  // K-subselect: lanes 0-15 hold K0,K1; 16-31 hold K2,K3
    const int myrow = half * 16 + m;

    for (int c = 0; c < NCHUNK; ++c) {
        const int cur = c & 1;
        __syncthreads();   // all waves done computing on the buffer we are about to refill
        if (c + 1 < NCHUNK) {
            const int nc = c + 1;
            #pragma unroll
            for (int it = 0; it < 4; ++it) {
                int idx = it * 256 + tid;
                int n   = idx >> 2;
                int c4  = idx & 3;
                async_ld_b128((uint32_t)(uintptr_t)&cbs[(nc & 1) * CBUF + n * CPAD + c4 * 4],
                              CB + n * D + nc * CHUNK + c4 * 4);
            }
            wait_async_le4();   // chunk c (and, on c==0, the X tile) has landed
        } else {
            wait_async_le0();
        }
        __syncthreads();

        const float* xrow  = &xs[myrow * XPAD + c * CHUNK + ks];
        const float* bbase = &cbs[cur * CBUF + (codehalf * 128 + m) * CPAD + ks];
        #pragma unroll
        for (int kk = 0; kk < 4; ++kk) {
            v2f a = *(const v2f*)(xrow + kk * 4);
            #pragma unroll
            for (int j = 0; j < 8; ++j) {
                v2f b = *(const v2f*)(bbase + j * 16 * CPAD + kk * 4);
                acc[j] = __builtin_amdgcn_wmma_f32_16x16x4_f32(
                    false, a, false, b, (short)0, acc[j], false, false);
            }
        }
    }

    // ---- per-token squared norms (xs is stable; needed only for argmin) ----
    if (tid < TOKPB) {
        float s = 0.f;
        for (int i = 0; i < D; ++i) { float v = xs[tid * XPAD + i]; s += v * v; }
        xnorm[tid] = s;
    }
    __syncthreads();

    // ---- distances + argmin ----
    // C layout: acc[j][r] = dot(token M, code N), M = r + 8*(lane>=16), N = codehalf*128 + j*16 + m
    float cbn_r[8];
    #pragma unroll
    for (int j = 0; j < 8; ++j) cbn_r[j] = cbn[codehalf * 128 + j * 16 + m];
    const int hi = lane >> 4;

    #pragma unroll
    for (int r = 0; r < 8; ++r) {
        int tok = half * 16 + r + hi * 8;
        float xn = xnorm[tok];
        float bestv = xn - 2.f * acc[0][r] + cbn_r[0];
        int   bestn = codehalf * 128 + m;
        #pragma unroll
        for (int j = 1; j < 8; ++j) {
            float dj = xn - 2.f * acc[j][r] + cbn_r[j];
            if (dj < bestv) { bestv = dj; bestn = codehalf * 128 + j * 16 + m; }
        }
        // reduce across the 16 lanes of this half-wave (xor masks < 16 keep halves separate)
        #pragma unroll
        for (int s = 1; s < 16; s <<= 1) {
            float ov = __shfl_xor(bestv, s, 32);
            int   on = __shfl_xor(bestn, s, 32);
            if (ov < bestv || (ov == bestv && on < bestn)) { bestv = ov; bestn = on; }
        }
        if (m == 0) {
            redv[codehalf * TOKPB + tok] = bestv;
            redi[codehalf * TOKPB + tok] = bestn;
        }
    }
    __syncthreads();

    if (tid < TOKPB) {
        float bv = redv[tid];
        int   bn = redi[tid];
        float v2 = redv[TOKPB + tid];
        int   n2 = redi[TOKPB + tid];
        if (v2 < bv || (v2 == bv && n2 < bn)) { bv = v2; bn = n2; }
        tokidx[tid] = bn;
        IDXF[tok0 + tid] = (float)bn;
    }
    __syncthreads();

    // ---- gather q = CB[idx] (codebook is L2-hot), write Q, accumulate commitment loss ----
    float lsum = 0.f;
    #pragma unroll
    for (int it = 0; it < 24; ++it) {
        int idx = it * 256 + tid;
        int row = idx / 96;
        int c4  = idx % 96;
        int code = tokidx[row];
        float4 cv = *(const float4*)(CB + code * D + c4 * 4);
        float4 xv = *(const float4*)&xs[row * XPAD + c4 * 4];
        *(float4*)(Q + (tok0 + row) * D + c4 * 4) = cv;
        float dx = cv.x - xv.x, dy = cv.y - xv.y, dz = cv.z - xv.z, dw = cv.w - xv.w;
        lsum += dx * dx + dy * dy + dz * dz + dw * dw;
    }
    #pragma unroll
    for (int s = 1; s < 32; s <<= 1) lsum += __shfl_xor(lsum, s, 32);
    if (lane == 0) lossred[wave] = lsum;
    __syncthreads();
    if (tid == 0) {
        float t = 0.f;
        #pragma unroll
        for (int w = 0; w < 8; ++w) t += lossred[w];
        atomicAdd(loss_out, t * beta_scale);
    }
}

extern "C" void kernel_launch(void* const* d_in, const int* in_sizes, int n_in,
                              void* d_out, int out_size, void* d_ws, size_t ws_size,
                              hipStream_t stream) {
    const float* l0  = (const float*)d_in[0];
    const float* l1  = (const float*)d_in[1];
    const float* l2  = (const float*)d_in[2];
    const float* cb0 = (const float*)d_in[3];
    const float* cb1 = (const float*)d_in[4];
    const float* cb2 = (const float*)d_in[5];

    const long n0 = in_sizes[0], n1 = in_sizes[1], n2 = in_sizes[2];
    const int  t0 = (int)(n0 / D), t1 = (int)(n1 / D), t2 = (int)(n2 / D);

    float* out  = (float*)d_out;
    float* q0   = out;
    float* q1   = q0 + n0;
    float* q2   = q1 + n1;
    float* loss = q2 + n2;
    float* i0   = loss + 1;
    float* i1   = i0 + t0;
    float* i2   = i1 + t1;

    float* norms = (float*)d_ws;   // 3 * 256 floats of ||cb||^2

    vq_norms_kernel<<<3, KCODES, 0, stream>>>(cb0, cb1, cb2, norms);
    vq_zero_loss_kernel<<<1, 1, 0, stream>>>(loss);

    vq_level_kernel<<<t0 / TOKPB, 256, 0, stream>>>(l0, cb0, norms + 0 * KCODES, q0, i0, loss,
                                                    0.05f / (float)n0);
    vq_level_kernel<<<t1 / TOKPB, 256, 0, stream>>>(l1, cb1, norms + 1 * KCODES, q1, i1, loss,
                                                    0.25f / (float)n1);
    vq_level_kernel<<<t2 / TOKPB, 256, 0, stream>>>(l2, cb2, norms + 2 * KCODES, q2, i2, loss,
                                                    0.60f / (float)n2);
}